// LinSinkhornPRModel_21371757265163
// MI455X (gfx1250) — compile-verified
//
#include <hip/hip_runtime.h>
#include <vector>

typedef __attribute__((ext_vector_type(2))) float v2f;
typedef __attribute__((ext_vector_type(8))) float v8f;

#define TPC 16          // 16x16 column tiles per chunk (256 columns)
#define NEG_BIG (-3.0e38f)

// ---------------------------------------------------------------------------
// out[M x 128] = X[M x 128] @ W^T   (W is 128x128, row-major)
// One wave per 16x16 output tile. f32 WMMA 16x16x4, K = 128 -> 32 chained.
// A layout: lane l holds A[l&15][2*(l>>4) + {0,1}]  (ISA 16x4 f32 A layout)
// B layout: lane l holds B[2*(l>>4)+{0,1}][l&15]    (ISA 4x16 f32 B layout)
// D layout: acc[v] = D[v + 8*(l>>4)][l&15]
// ---------------------------------------------------------------------------
__global__ void gemm_xwt(const float* __restrict__ X, const float* __restrict__ W,
                         float* __restrict__ out, int M) {
  int wave = (blockIdx.x * blockDim.x + threadIdx.x) >> 5;
  int lane = threadIdx.x & 31;
  int half = lane >> 4;
  int lmod = lane & 15;
  int it = wave >> 3;       // 8 column tiles (128/16)
  int nt = wave & 7;
  if (it * 16 >= M) return; // wave-uniform
  int i0 = it * 16, n0 = nt * 16;

  const float* arow = X + (size_t)(i0 + lmod) * 128 + 2 * half;
  const float* brow = W + (size_t)(n0 + lmod) * 128 + 2 * half; // B[k][n] = W[n][k]

  v8f acc = {};
#pragma unroll
  for (int k = 0; k < 32; ++k) {
    v2f a = *(const v2f*)(arow + k * 4);
    v2f b = *(const v2f*)(brow + k * 4);
    acc = __builtin_amdgcn_wmma_f32_16x16x4_f32(false, a, false, b, (short)0, acc,
                                                false, false);
  }
#pragma unroll
  for (int v = 0; v < 8; ++v)
    out[(size_t)(i0 + v + 8 * half) * 128 + n0 + lmod] = acc[v];
}

// ---------------------------------------------------------------------------
// Fused cost + online-logsumexp partial pass:
//   for rows i in this wave's 16-row tile, over a chunk of columns j:
//     t_ij = (hlogY[j] + g[j]/eps) - C_ij/eps,  C_ij = 0.5*max(sx+sy-2*x.y, 0)
//   accumulate running (max, sumexp) per row; write per-chunk partials.
// ---------------------------------------------------------------------------
__global__ void softmin_partial(const float* __restrict__ X,
                                const float* __restrict__ sqX, int nX,
                                const float* __restrict__ Y,
                                const float* __restrict__ sqY,
                                const float* __restrict__ hlogY,
                                const float* __restrict__ g, float inv_eps,
                                int nYtiles, float2* __restrict__ partials) {
  int wave = (blockIdx.x * blockDim.x + threadIdx.x) >> 5;
  int lane = threadIdx.x & 31;
  int half = lane >> 4;
  int lmod = lane & 15;
  int i0 = wave * 16;
  int chunk = blockIdx.y;
  int jt0 = chunk * TPC;
  int jt1 = jt0 + TPC < nYtiles ? jt0 + TPC : nYtiles;

  // Preload A tile for full K=128 (reused across all column tiles)
  const float* arow = X + (size_t)(i0 + lmod) * 128 + 2 * half;
  v2f a[32];
#pragma unroll
  for (int k = 0; k < 32; ++k) a[k] = *(const v2f*)(arow + k * 4);

  float sqx[8];
#pragma unroll
  for (int v = 0; v < 8; ++v) sqx[v] = sqX[i0 + v + 8 * half];

  float mx[8], s[8];
#pragma unroll
  for (int v = 0; v < 8; ++v) { mx[v] = NEG_BIG; s[v] = 0.0f; }

  for (int jt = jt0; jt < jt1; ++jt) {
    int j0 = jt * 16;
    const float* brow = Y + (size_t)(j0 + lmod) * 128 + 2 * half;
    v8f acc = {};
#pragma unroll
    for (int k = 0; k < 32; ++k) {
      v2f b = *(const v2f*)(brow + k * 4);
      acc = __builtin_amdgcn_wmma_f32_16x16x4_f32(false, a[k], false, b, (short)0,
                                                  acc, false, false);
    }
    float sqy = sqY[j0 + lmod];
    float w = hlogY[j0 + lmod];
    if (g) w += g[j0 + lmod] * inv_eps;
#pragma unroll
    for (int v = 0; v < 8; ++v) {
      float c = 0.5f * fmaxf(sqx[v] + sqy - 2.0f * acc[v], 0.0f);
      float t = w - c * inv_eps;
      float m2 = fmaxf(mx[v], t);
      s[v] = s[v] * __expf(mx[v] - m2) + __expf(t - m2);
      mx[v] = m2;
    }
  }

  // merge (max,sum) across the 16 lanes of each half-wave
#pragma unroll
  for (int off = 1; off < 16; off <<= 1) {
#pragma unroll
    for (int v = 0; v < 8; ++v) {
      float mo = __shfl_xor(mx[v], off, 32);
      float so = __shfl_xor(s[v], off, 32);
      float m2 = fmaxf(mx[v], mo);
      s[v] = s[v] * __expf(mx[v] - m2) + so * __expf(mo - m2);
      mx[v] = m2;
    }
  }
  if (lmod == 0) {
#pragma unroll
    for (int v = 0; v < 8; ++v)
      partials[(size_t)chunk * nX + i0 + v + 8 * half] = make_float2(mx[v], s[v]);
  }
}

// Merge chunk partials -> f_new = -eps*(log S + M); optionally 0.5*(fprev+f_new)
__global__ void combine_kernel(const float2* __restrict__ partials, int nchunks,
                               int n, float eps, const float* __restrict__ fprev,
                               float* __restrict__ fout, int avg) {
  int i = blockIdx.x * blockDim.x + threadIdx.x;
  if (i >= n) return;
  float M = NEG_BIG, S = 0.0f;
  for (int c = 0; c < nchunks; ++c) {
    float2 p = partials[(size_t)c * n + i];
    float m2 = fmaxf(M, p.x);
    S = S * __expf(M - m2) + p.y * __expf(p.x - m2);
    M = m2;
  }
  float ft = -eps * (logf(S) + M);
  fout[i] = avg ? 0.5f * (fprev[i] + ft) : ft;
}

__global__ void rowsq_kernel(const float* __restrict__ X, float* __restrict__ sq,
                             int rows) {
  int wave = (blockIdx.x * blockDim.x + threadIdx.x) >> 5;
  int lane = threadIdx.x & 31;
  if (wave >= rows) return;
  const float* r = X + (size_t)wave * 128;
  float acc = 0.0f;
#pragma unroll
  for (int k = 0; k < 4; ++k) {
    float x = r[lane + 32 * k];
    acc += x * x;
  }
#pragma unroll
  for (int off = 16; off > 0; off >>= 1) acc += __shfl_xor(acc, off, 32);
  if (lane == 0) sq[wave] = acc;
}

__global__ void log_kernel(const float* __restrict__ in, float* __restrict__ out,
                           int n) {
  int i = blockIdx.x * blockDim.x + threadIdx.x;
  if (i < n) out[i] = logf(in[i]);
}

__global__ void dist_reduce_kernel(const float* __restrict__ a,
                                   const float* __restrict__ f1,
                                   const float* __restrict__ f2, int n,
                                   const float* __restrict__ b,
                                   const float* __restrict__ g1,
                                   const float* __restrict__ g2, int m,
                                   float* __restrict__ out) {
  __shared__ float red[256];
  int tid = threadIdx.x;
  float acc = 0.0f;
  for (int i = tid; i < n; i += 256) acc += a[i] * (f1[i] - f2[i]);
  for (int j = tid; j < m; j += 256) acc += b[j] * (g1[j] - g2[j]);
  red[tid] = acc;
  __syncthreads();
  for (int off = 128; off > 0; off >>= 1) {
    if (tid < off) red[tid] += red[tid + off];
    __syncthreads();
  }
  if (tid == 0) out[0] = red[0];
}

__global__ void sigmoid_kernel(const float* __restrict__ d1,
                               const float* __restrict__ d2,
                               float* __restrict__ out) {
  if (threadIdx.x == 0 && blockIdx.x == 0) {
    float z = d2[0] - d1[0];  // SCALING_FACTOR = 1.0
    out[0] = 1.0f / (1.0f + expf(-z));
  }
}

// ---------------------------------------------------------------------------
extern "C" void kernel_launch(void* const* d_in, const int* in_sizes, int n_in,
                              void* d_out, int out_size, void* d_ws, size_t ws_size,
                              hipStream_t stream) {
  const float* d  = (const float*)d_in[0];
  const float* si = (const float*)d_in[1];
  const float* sj = (const float*)d_in[2];
  const float* h  = (const float*)d_in[3];
  const float* hi = (const float*)d_in[4];
  const float* hj = (const float*)d_in[5];
  const float* W  = (const float*)d_in[6];

  const int N = in_sizes[3];  // 4096
  const int M = in_sizes[4];  // 2048

  // ---- workspace carve-up (all counts even -> 8B alignment preserved) ----
  float* p = (float*)d_ws;
  auto alloc = [&](size_t n) { float* r = p; p += n; return r; };
  float* td   = alloc((size_t)N * 128);
  float* tsi  = alloc((size_t)M * 128);
  float* tsj  = alloc((size_t)M * 128);
  float* sqd  = alloc(N);
  float* sqi  = alloc(M);
  float* sqj  = alloc(M);
  float* lgh  = alloc(N);
  float* lgi  = alloc(M);
  float* lgj  = alloc(M);
  float* f_ab = alloc(N);
  float* g_ba = alloc(M);
  float* f_aa = alloc(N);
  float* g_bb = alloc(M);
  float* f_ab_f = alloc(N);
  float* g_ba_f = alloc(M);
  float* f_aa_f = alloc(N);
  float* g_bb_f = alloc(M);
  float2* pf  = (float2*)alloc((size_t)2 * 16 * N);
  float2* pg  = (float2*)alloc((size_t)2 * 16 * N);
  float2* pfa = (float2*)alloc((size_t)2 * 16 * N);
  float2* pgb = (float2*)alloc((size_t)2 * 16 * N);
  float* dist1 = alloc(2);
  float* dist2 = dist1 + 1;

  // ---- transform: X @ W^T ----
  auto launch_gemm = [&](const float* X, float* out, int rows) {
    int waves = (rows / 16) * 8;
    gemm_xwt<<<dim3(waves / 4), dim3(128), 0, stream>>>(X, W, out, rows);
  };
  launch_gemm(d, td, N);
  launch_gemm(si, tsi, M);
  launch_gemm(sj, tsj, M);

  rowsq_kernel<<<dim3(N / 8), dim3(256), 0, stream>>>(td, sqd, N);
  rowsq_kernel<<<dim3(M / 8), dim3(256), 0, stream>>>(tsi, sqi, M);
  rowsq_kernel<<<dim3(M / 8), dim3(256), 0, stream>>>(tsj, sqj, M);
  log_kernel<<<dim3((N + 255) / 256), dim3(256), 0, stream>>>(h, lgh, N);
  log_kernel<<<dim3((M + 255) / 256), dim3(256), 0, stream>>>(hi, lgi, M);
  log_kernel<<<dim3((M + 255) / 256), dim3(256), 0, stream>>>(hj, lgj, M);

  // ---- epsilon schedule (geomloss annealing) ----
  std::vector<float> eps_list;
  {
    double sigma = 32.0;
    while (sigma > 0.05) {
      eps_list.push_back((float)(sigma * sigma));
      sigma *= 0.5;
    }
    eps_list.push_back((float)(0.05 * 0.05));
  }
  const float eps_f = (float)(0.05 * 0.05);

  auto softmin = [&](const float* X, const float* sqX, int nX, const float* Y,
                     const float* sqY, int nY, const float* blog, const float* gY,
                     float eps, float2* part, int* nch_out) {
    int nYt = nY / 16;
    int nch = (nYt + TPC - 1) / TPC;
    dim3 grid((nX / 16) / 4, nch);
    softmin_partial<<<grid, dim3(128), 0, stream>>>(X, sqX, nX, Y, sqY, blog, gY,
                                                    1.0f / eps, nYt, part);
    *nch_out = nch;
  };
  auto combine = [&](float2* part, int nch, int n, float eps, const float* fprev,
                     float* fout, int avg) {
    combine_kernel<<<dim3((n + 255) / 256), dim3(256), 0, stream>>>(
        part, nch, n, eps, fprev, fout, avg);
  };

  auto sinkhorn = [&](const float* aw, const float* alog, const float* x,
                      const float* sqx, int n, const float* bw, const float* blog,
                      const float* y, const float* sqy, int m, float* dist) {
    int c1, c2, c3, c4;
    float e0 = eps_list[0];
    // init potentials
    softmin(x, sqx, n, y, sqy, m, blog, nullptr, e0, pf, &c1);
    combine(pf, c1, n, e0, nullptr, f_ab, 0);
    softmin(y, sqy, m, x, sqx, n, alog, nullptr, e0, pg, &c2);
    combine(pg, c2, m, e0, nullptr, g_ba, 0);
    softmin(x, sqx, n, x, sqx, n, alog, nullptr, e0, pfa, &c3);
    combine(pfa, c3, n, e0, nullptr, f_aa, 0);
    softmin(y, sqy, m, y, sqy, m, blog, nullptr, e0, pgb, &c4);
    combine(pgb, c4, m, e0, nullptr, g_bb, 0);

    for (size_t e = 0; e < eps_list.size(); ++e) {
      float eps = eps_list[e];
      // all four softmins read the OLD potentials ...
      softmin(x, sqx, n, y, sqy, m, blog, g_ba, eps, pf, &c1);
      softmin(y, sqy, m, x, sqx, n, alog, f_ab, eps, pg, &c2);
      softmin(x, sqx, n, x, sqx, n, alog, f_aa, eps, pfa, &c3);
      softmin(y, sqy, m, y, sqy, m, blog, g_bb, eps, pgb, &c4);
      // ... then the averaged updates land
      combine(pf, c1, n, eps, f_ab, f_ab, 1);
      combine(pg, c2, m, eps, g_ba, g_ba, 1);
      combine(pfa, c3, n, eps, f_aa, f_aa, 1);
      combine(pgb, c4, m, eps, g_bb, g_bb, 1);
    }

    // final extrapolation at blur^2
    softmin(x, sqx, n, y, sqy, m, blog, g_ba, eps_f, pf, &c1);
    softmin(y, sqy, m, x, sqx, n, alog, f_ab, eps_f, pg, &c2);
    softmin(x, sqx, n, x, sqx, n, alog, f_aa, eps_f, pfa, &c3);
    softmin(y, sqy, m, y, sqy, m, blog, g_bb, eps_f, pgb, &c4);
    combine(pf, c1, n, eps_f, nullptr, f_ab_f, 0);
    combine(pg, c2, m, eps_f, nullptr, g_ba_f, 0);
    combine(pfa, c3, n, eps_f, nullptr, f_aa_f, 0);
    combine(pgb, c4, m, eps_f, nullptr, g_bb_f, 0);

    dist_reduce_kernel<<<dim3(1), dim3(256), 0, stream>>>(
        aw, f_ab_f, f_aa_f, n, bw, g_ba_f, g_bb_f, m, dist);
  };

  sinkhorn(h, lgh, td, sqd, N, hi, lgi, tsi, sqi, M, dist1);
  sinkhorn(h, lgh, td, sqd, N, hj, lgj, tsj, sqj, M, dist2);
  sigmoid_kernel<<<dim3(1), dim3(32), 0, stream>>>(dist1, dist2, (float*)d_out);
}